// SoftConstrainedIPMModule_36704790512292
// MI455X (gfx1250) — compile-verified
//
#include <hip/hip_runtime.h>

typedef float v2f __attribute__((ext_vector_type(2)));
typedef float v8f __attribute__((ext_vector_type(8)));

namespace {
constexpr int   kN        = 12;    // variables per QP
constexpr int   kM        = 11;    // constraints per QP
constexpr int   kW        = 1024;  // problems per batch row
constexpr float kBeta2    = 0.055f;
constexpr float kEps      = 0.01f;
constexpr int   kLdsStride = 13;   // 12 + 1 pad (odd stride -> conflict-free)
}

__device__ __forceinline__ float hash_u01(unsigned int x) {
  x ^= x >> 17; x *= 0xed5ad4bbu;
  x ^= x >> 11; x *= 0xac4c1b51u;
  x ^= x >> 15; x *= 0x31848babu;
  x ^= x >> 14;
  return (float)(x >> 8) * (1.0f / 16777216.0f);   // [0,1)
}

// || [R1; R2] ||_2 at (S, lam) with fixed barrier parameter 1/t.
__device__ __forceinline__ float resid_norm(const float S[kN], const float lam[kM],
                                            const float dv[kN], const float bv[kN],
                                            float inv_t) {
  float acc = 0.0f;
#pragma unroll
  for (int j = 0; j < kN; ++j) {
    float r1 = dv[j] * S[j] + bv[j];
    if (j > 0)  r1 += -4.0f * S[j - 1] - lam[j - 1];
    if (j < kM) r1 += lam[j];
    acc += r1 * r1;
  }
#pragma unroll
  for (int i = 0; i < kM; ++i) {
    float a  = S[i] - S[i + 1];
    float r2 = -lam[i] * a - inv_t;
    acc += r2 * r2;
  }
  return sqrtf(acc);
}

__global__ __launch_bounds__(256) void ipm_qp_kernel(
    const float* __restrict__ c_lambda_p,
    const float* __restrict__ Mu,     // (B, N, W)
    const float* __restrict__ Sig2,   // (B, N, W)
    const float* __restrict__ R,      // (B, N, W)
    float* __restrict__ Out)          // (B, N, W)
{
  __shared__ float sS[256 * kLdsStride];

  const int tid = threadIdx.x;
  const int p   = blockIdx.x * 256 + tid;     // problem id, 0..16383 (exact)
  const int b   = p >> 10;
  const int w   = p & (kW - 1);
  const float c_lambda = c_lambda_p[0];
  const size_t base = (size_t)b * kN * kW + (size_t)w;

  // ---- load per-problem vectors (coalesced across lanes: consecutive w) ----
  float mu[kN], s2[kN], rr[kN];
#pragma unroll
  for (int j = 0; j < kN; ++j) {
    mu[j] = Mu[base + (size_t)j * kW];
    s2[j] = Sig2[base + (size_t)j * kW];
    rr[j] = R[base + (size_t)j * kW];
  }

  // ---- build H diag / b vector (H is diag + subdiagonal(-4)) ----
  float dv[kN], bv[kN];
#pragma unroll
  for (int j = 0; j < kN; ++j) {
    dv[j] = c_lambda / s2[j] + 2.0f + ((j < kN - 1) ? 2.0f : 0.0f);
    bv[j] = -c_lambda * mu[j] / s2[j] - 2.0f * rr[j]
          + ((j < kN - 1) ? 2.0f * rr[j + 1] : 0.0f);
  }

  // ---- S0 = sort(mu) ascending: fully-unrolled sorting network (constant idx) ----
  float S[kN];
#pragma unroll
  for (int j = 0; j < kN; ++j) S[j] = mu[j];
#pragma unroll
  for (int i = 0; i < kN - 1; ++i)
#pragma unroll
    for (int j = 0; j < kN - 1 - i; ++j) {
      float lo = fminf(S[j], S[j + 1]);
      float hi = fmaxf(S[j], S[j + 1]);
      S[j] = lo; S[j + 1] = hi;
    }

  // ---- Lam0 in (0,20), beta3 in (10,11): deterministic hash uniforms ----
  float lam[kM];
#pragma unroll
  for (int i = 0; i < kM; ++i)
    lam[i] = 20.0f * hash_u01(((unsigned)p * 31u + (unsigned)i) * 2654435761u + 0x9e3779b9u)
           + 1e-4f;  // keep strictly interior
  const float beta3 = 10.0f + hash_u01((unsigned)p * 2246822519u + 0x85ebca6bu);

  // =====================================================================
  // AS0 = A @ S0 via V_WMMA_F32_16X16X4_F32.  A (11x12, padded 16x12) is
  // the SAME matrix for every problem, so this is a genuine shared-operand
  // GEMM: (16x12) x (12 x 16 problems), K split 4+4+4. Two problem batches
  // of 16 per wave. Uniform control flow here -> EXEC all ones.
  // =====================================================================
#pragma unroll
  for (int j = 0; j < kN; ++j) sS[tid * kLdsStride + j] = S[j];
  __syncthreads();

  const int lane      = tid & 31;
  const int waveStart = tid & ~31;          // first row of this wave in LDS
  const int mrow      = lane & 15;          // M index this lane supplies for A
  const int koff      = (lane < 16) ? 0 : 2;

  v8f acc0 = {};
  v8f acc1 = {};
#pragma unroll
  for (int kk = 0; kk < kN; kk += 4) {
    // A operand: A[m][k] = (k==m) - (k==m+1), rows >= 11 are zero padding.
    v2f av;
    {
      int k0 = kk + koff, k1 = kk + koff + 1;
      float a0 = 0.0f, a1 = 0.0f;
      if (mrow < kM) {
        a0 = (k0 == mrow) ? 1.0f : ((k0 == mrow + 1) ? -1.0f : 0.0f);
        a1 = (k1 == mrow) ? 1.0f : ((k1 == mrow + 1) ? -1.0f : 0.0f);
      }
      av.x = a0; av.y = a1;
    }
    // B operands: column n = problem (lane&15) of each 16-problem batch.
    int row0 = (waveStart + (lane & 15)) * kLdsStride;
    int row1 = row0 + 16 * kLdsStride;
    v2f b0, b1;
    b0.x = sS[row0 + kk + koff];  b0.y = sS[row0 + kk + koff + 1];
    b1.x = sS[row1 + kk + koff];  b1.y = sS[row1 + kk + koff + 1];

    acc0 = __builtin_amdgcn_wmma_f32_16x16x4_f32(false, av, false, b0,
                                                 (short)0, acc0, false, false);
    acc1 = __builtin_amdgcn_wmma_f32_16x16x4_f32(false, av, false, b1,
                                                 (short)0, acc1, false, false);
  }
  __syncthreads();
  // Scatter D back: vgpr r of lane L holds AS0[m][n], m = r + (L<16?0:8),
  // n = L&15 within the batch.
  {
    int mbase = (lane < 16) ? 0 : 8;
    int n = lane & 15;
#pragma unroll
    for (int r = 0; r < 8; ++r) {
      int m = mbase + r;
      if (m < kM) {
        sS[(waveStart + n) * kLdsStride + m]      = acc0[r];
        sS[(waveStart + 16 + n) * kLdsStride + m] = acc1[r];
      }
    }
  }
  __syncthreads();
  float as[kM];
#pragma unroll
  for (int i = 0; i < kM; ++i) as[i] = sS[tid * kLdsStride + i];

  // =====================================================================
  // Primal-dual IPM, one problem per lane, all state in VGPRs.
  // The K=23 Newton system is solved exactly via its tridiagonal 12x12
  // Schur complement  (H - A^T diag(lam/as) A) dS = -R1 - A^T (R2/as).
  // =====================================================================
  float metric = 1e30f;
  int it = 0;
  while (metric >= kEps && it <= 100) {
    // t = -beta3*M / <as, lam>;  only 1/t is needed.
    float dot = 0.0f;
#pragma unroll
    for (int i = 0; i < kM; ++i) dot += as[i] * lam[i];
    const float inv_t = -dot / (beta3 * (float)kM);

    // Residual at (S, lam) and its norm.
    float r1[kN], r2[kM];
    float rn2 = 0.0f;
#pragma unroll
    for (int j = 0; j < kN; ++j) {
      float v = dv[j] * S[j] + bv[j];
      if (j > 0)  v += -4.0f * S[j - 1] - lam[j - 1];
      if (j < kM) v += lam[j];
      r1[j] = v; rn2 += v * v;
    }
#pragma unroll
    for (int i = 0; i < kM; ++i) {
      float v = -lam[i] * as[i] - inv_t;
      r2[i] = v; rn2 += v * v;
    }
    const float RNorm = sqrtf(rn2);

    // Schur complement (tridiagonal) + Thomas solve.
    float g[kM], h[kM];
#pragma unroll
    for (int i = 0; i < kM; ++i) {
      float asafe = fminf(as[i], -1e-20f);   // as <= 0 in the interior
      g[i] = lam[i] / asafe;
      h[i] = r2[i] / asafe;
    }
    float cp[kN], dp[kN];
    {
      float dg0  = dv[0] - g[0];
      float rhs0 = -r1[0] - h[0];
      cp[0] = g[0] / dg0;
      dp[0] = rhs0 / dg0;
#pragma unroll
      for (int j = 1; j < kN; ++j) {
        float lj  = -4.0f + g[j - 1];
        float dgj = dv[j] - ((j < kM ? g[j] : 0.0f) + g[j - 1]);
        float rhs = -r1[j] - ((j < kM ? h[j] : 0.0f) - h[j - 1]);
        float den = dgj - lj * cp[j - 1];
        cp[j] = (j < kN - 1 ? g[j] : 0.0f) / den;
        dp[j] = (rhs - lj * dp[j - 1]) / den;
      }
    }
    float pds[kN];
    pds[kN - 1] = dp[kN - 1];
#pragma unroll
    for (int j = kN - 2; j >= 0; --j) pds[j] = dp[j] - cp[j] * pds[j + 1];

    float pdl[kM];
#pragma unroll
    for (int i = 0; i < kM; ++i) {
      float asafe = fminf(as[i], -1e-20f);
      pdl[i] = (r2[i] - lam[i] * (pds[i] - pds[i + 1])) / asafe;
    }

    // Fraction-to-boundary step (alpha_i = 1 when pdl >= 0, incl. NaN->1 rule).
    float alpha = 1e30f;
#pragma unroll
    for (int i = 0; i < kM; ++i) {
      float ai = (pdl[i] < 0.0f) ? (-lam[i] / pdl[i]) : 1.0f;
      alpha = fminf(alpha, ai);
    }
    alpha *= 0.99f;

    float Sp[kN], Lp[kM];
#pragma unroll
    for (int j = 0; j < kN; ++j) Sp[j] = S[j];
#pragma unroll
    for (int i = 0; i < kM; ++i) Lp[i] = lam[i];

#pragma unroll
    for (int j = 0; j < kN; ++j) S[j] = Sp[j] + alpha * pds[j];
#pragma unroll
    for (int i = 0; i < kM; ++i) as[i] = S[i] - S[i + 1];

    // Feasibility backtracking: halve alpha while any (A S > d).
    for (int fi = 0; fi < 100; ++fi) {
      bool infeas = false;
#pragma unroll
      for (int i = 0; i < kM; ++i) infeas = infeas || (as[i] > 0.0f);
      if (!infeas) break;
      alpha *= 0.5f;
#pragma unroll
      for (int j = 0; j < kN; ++j) S[j] = Sp[j] + alpha * pds[j];
#pragma unroll
      for (int i = 0; i < kM; ++i) as[i] = S[i] - S[i + 1];
    }

#pragma unroll
    for (int i = 0; i < kM; ++i) lam[i] = Lp[i] + alpha * pdl[i];

    // Merit line search: R2N <= (1 - beta2*alpha) * RNorm.
    float R2N = resid_norm(S, lam, dv, bv, inv_t);
    for (int li = 0; li < 100; ++li) {
      if (!(R2N > (1.0f - kBeta2 * alpha) * RNorm)) break;
      alpha *= 0.5f;
#pragma unroll
      for (int j = 0; j < kN; ++j) S[j] = Sp[j] + alpha * pds[j];
#pragma unroll
      for (int i = 0; i < kM; ++i) lam[i] = Lp[i] + alpha * pdl[i];
      R2N = resid_norm(S, lam, dv, bv, inv_t);
    }
#pragma unroll
    for (int i = 0; i < kM; ++i) as[i] = S[i] - S[i + 1];

    metric = R2N;
    ++it;
  }

  // Output layout (B, N, W): same striding as inputs -> coalesced stores.
#pragma unroll
  for (int j = 0; j < kN; ++j) Out[base + (size_t)j * kW] = S[j];
}

extern "C" void kernel_launch(void* const* d_in, const int* in_sizes, int n_in,
                              void* d_out, int out_size, void* d_ws, size_t ws_size,
                              hipStream_t stream) {
  (void)in_sizes; (void)n_in; (void)out_size; (void)d_ws; (void)ws_size;
  const float* c_lambda = (const float*)d_in[0];
  const float* Mu       = (const float*)d_in[1];
  const float* Sig2     = (const float*)d_in[2];
  const float* R        = (const float*)d_in[3];
  float* Out            = (float*)d_out;

  const int problems = 16 * 1024;            // B * W
  dim3 grid(problems / 256), block(256);
  ipm_qp_kernel<<<grid, block, 0, stream>>>(c_lambda, Mu, Sig2, R, Out);
}